// WrapMSELoss_64115271794917
// MI455X (gfx1250) — compile-verified
//
#include <hip/hip_runtime.h>
#include <hip/hip_bf16.h>

typedef __attribute__((ext_vector_type(2))) float v2f;
typedef __attribute__((ext_vector_type(4))) float v4f;
typedef __attribute__((ext_vector_type(8))) float v8f;

#define BS 16
#define NC 19
#define HH 384
#define WW 640
#define NPIX (HH * WW)                 // 245760 per channel
#define NCH (BS * NC)                  // 304 channels
#define GX 30                          // blocks along pixel dim
#define ITERS 8                        // 30*256*4*8 == 245760 exactly
#define NPART (NCH * GX)               // 9120 partials
#define NTOT 74711040.0f               // 16*19*384*640

__device__ __forceinline__ float fast_exp2(float x) {
#if __has_builtin(__builtin_amdgcn_exp2f)
  return __builtin_amdgcn_exp2f(x);
#else
  return exp2f(x);
#endif
}

__device__ __forceinline__ float fast_sigmoid(float x) {
#if __has_builtin(__builtin_amdgcn_tanhf)
  // gfx1250 hardware tanh: one transcendental instead of exp+rcp
  return fmaf(__builtin_amdgcn_tanhf(x * 0.5f), 0.5f, 0.5f);
#else
  float e = fast_exp2(x * -1.44269504f);
#if __has_builtin(__builtin_amdgcn_rcpf)
  return __builtin_amdgcn_rcpf(1.0f + e);
#else
  return 1.0f / (1.0f + e);
#endif
#endif
}

// ---- kernel 1: invert class_labels into map[b*NC + label] = k (last wins) ----
__global__ void build_map_kernel(const int* __restrict__ labels, int* __restrict__ map) {
  int t = threadIdx.x;
  if (t < NCH) map[t] = -1;
  __syncthreads();
  if (t < NCH) {
    int b = t / NC;
    int k = t - b * NC;
    int lab = labels[b * NC + k];
    if (lab >= 0 && lab < NC) atomicMax(&map[b * NC + lab], k);
  }
}

// ---- kernel 2: streaming sigmoid/gaussian MSE with WMMA f32 accumulation ----
__global__ __launch_bounds__(256) void mse_wmma_kernel(const float* __restrict__ pred,
                                                       const float* __restrict__ kp,
                                                       const int* __restrict__ map,
                                                       float* __restrict__ partials) {
  const int ch = blockIdx.y;          // b*NC + c
  const int b = ch / NC;
  const int k = map[ch];
  float x0 = -1.0e6f, y0 = -1.0e6f;   // invalid channel -> gaussian never fires
  if (k >= 0) {
    x0 = kp[(b * NC + k) * 2 + 0];
    y0 = kp[(b * NC + k) * 2 + 1];
  }
  const float* p = pred + (size_t)ch * NPIX;
  const int tid = threadIdx.x;

  v2f bones;
  bones[0] = 1.0f;
  bones[1] = 1.0f;
  v8f acc = {};

  #pragma unroll
  for (int it = 0; it < ITERS; ++it) {
    int px = (((it * GX) + (int)blockIdx.x) * 256 + tid) * 4;
    v4f v = __builtin_nontemporal_load((const v4f*)(p + px));
    int y = px / WW;
    int xb = px - y * WW;
    float fy = (float)y - y0;
    float dy2 = fy * fy;
    float fxb = (float)xb - x0;       // one cvt per 4 pixels

    float d2v[4];
    #pragma unroll
    for (int j = 0; j < 4; ++j) {
      float fx = fxb + (float)j;
      d2v[j] = fmaf(fx, fx, dy2);
    }
    float dmin = fminf(fminf(d2v[0], d2v[1]), fminf(d2v[2], d2v[3]));

    // Wave-uniform branch: ballot-derived scalar condition -> s_cbranch, so
    // the ~99% of waves far from the keypoint skip all gaussian v_exp work.
    float g[4] = {0.0f, 0.0f, 0.0f, 0.0f};
    if (__any(dmin < 400.0f)) {
      #pragma unroll
      for (int j = 0; j < 4; ++j) {
        float e = fast_exp2(d2v[j] * -0.1803368801f);  // exp(-d2/8)
        g[j] = (d2v[j] < 400.0f) ? e : 0.0f;
      }
    }

    float pv[4] = {v[0], v[1], v[2], v[3]};
    float sq[4];
    #pragma unroll
    for (int j = 0; j < 4; ++j) {
      float diff = fast_sigmoid(pv[j]) - g[j];
      sq[j] = diff * diff;
    }

    // Pack 2 squared diffs per lane into the 16x4 f32 A-tile; B = ones.
    // D[m][n] += row_sum(A): every column replicates the sum -> 16x scale.
    v2f a0; a0[0] = sq[0]; a0[1] = sq[1];
    v2f a1; a1[0] = sq[2]; a1[1] = sq[3];
    acc = __builtin_amdgcn_wmma_f32_16x16x4_f32(false, a0, false, bones,
                                                (short)0, acc, false, false);
    acc = __builtin_amdgcn_wmma_f32_16x16x4_f32(false, a1, false, bones,
                                                (short)0, acc, false, false);
  }

  float vsum = acc[0] + acc[1] + acc[2] + acc[3] + acc[4] + acc[5] + acc[6] + acc[7];
  #pragma unroll
  for (int off = 16; off >= 1; off >>= 1)
    vsum += __shfl_xor(vsum, off, 32);

  __shared__ float lds[8];
  int wave = tid >> 5;
  int lane = tid & 31;
  if (lane == 0) lds[wave] = vsum;
  __syncthreads();
  if (tid == 0) {
    float t = 0.0f;
    #pragma unroll
    for (int i = 0; i < 8; ++i) t += lds[i];
    partials[blockIdx.y * GX + blockIdx.x] = t * 0.0625f;  // undo 16x WMMA replication
  }
}

// ---- kernel 3: deterministic final reduction of 9120 partials ----
__global__ __launch_bounds__(256) void final_reduce_kernel(const float* __restrict__ partials,
                                                           float* __restrict__ out) {
  __shared__ float lds[256];
  float s = 0.0f;
  for (int i = threadIdx.x; i < NPART; i += 256) s += partials[i];
  lds[threadIdx.x] = s;
  __syncthreads();
  for (int o = 128; o > 0; o >>= 1) {
    if (threadIdx.x < (unsigned)o) lds[threadIdx.x] += lds[threadIdx.x + o];
    __syncthreads();
  }
  if (threadIdx.x == 0) out[0] = lds[0] * (1.0f / NTOT);
}

extern "C" void kernel_launch(void* const* d_in, const int* in_sizes, int n_in,
                              void* d_out, int out_size, void* d_ws, size_t ws_size,
                              hipStream_t stream) {
  const float* pred = (const float*)d_in[0];
  const float* keypoints = (const float*)d_in[1];
  const int* class_labels = (const int*)d_in[2];
  float* out = (float*)d_out;

  float* partials = (float*)d_ws;                          // 9120 floats
  int* map = (int*)((char*)d_ws + NPART * sizeof(float));  // 304 ints

  build_map_kernel<<<1, 320, 0, stream>>>(class_labels, map);
  mse_wmma_kernel<<<dim3(GX, NCH), 256, 0, stream>>>(pred, keypoints, map, partials);
  final_reduce_kernel<<<1, 256, 0, stream>>>(partials, out);
}